// Conv2d_NN_44976897523813
// MI455X (gfx1250) — compile-verified
//
#include <hip/hip_runtime.h>
#include <math.h>

typedef __attribute__((ext_vector_type(2))) float v2f;
typedef __attribute__((ext_vector_type(8))) float v8f;

#define BATCH 32
#define CIN   32
#define COUT  32
#define HH    64
#define WW    64
#define C1    136      // (CIN+2)*4  (pixel-unshuffled channels)
#define C1P   144      // C1 padded to multiple of 16 for WMMA O-tiles
#define NTOK  1024     // 32*32 tokens after unshuffle
#define WS_   32
#define KNN   9
#define CK    1224     // C1*KNN contraction length of neighbor conv

// ---------------------------------------------------------------------------
// Kernel A: build x2 in token-major AND channel-major layouts + per-token sq
// x2[c1, n] ; c1 = c*4 + ri*2 + rj ; value = concat(x, coords)[c][2hs+ri][2ws+rj]
// ---------------------------------------------------------------------------
__global__ __launch_bounds__(256)
void prep_kernel(const float* __restrict__ xin,
                 float* __restrict__ x2tok,    // [B][N][C1]
                 float* __restrict__ x2chan,   // [B][C1][N]
                 float* __restrict__ sq) {     // [B][N]
  int n = blockIdx.x * blockDim.x + threadIdx.x;   // 0..1023
  int b = blockIdx.y;
  int hs = n >> 5, ws = n & 31;
  float acc = 0.f;
  for (int c1 = 0; c1 < C1; ++c1) {
    int c = c1 >> 2, sub = c1 & 3;
    int ri = sub >> 1, rj = sub & 1;
    int h = 2 * hs + ri, w = 2 * ws + rj;
    float v;
    if (c < CIN) {
      v = xin[(((size_t)(b * CIN + c)) * HH + h) * WW + w];
    } else {
      float fh = (float)h, fw = (float)w;
      float nrm = fmaxf(sqrtf(fh * fh + fw * fw), 1e-12f);
      v = (c == CIN) ? (fh / nrm) : (fw / nrm);
    }
    x2tok [((size_t)(b * NTOK + n)) * C1 + c1]  = v;
    x2chan[((size_t)(b * C1 + c1)) * NTOK + n]  = v;
    acc += v * v;
  }
  sq[b * NTOK + n] = acc;
}

// ---------------------------------------------------------------------------
// Kernel A2: transpose weights w1[o][ck] -> wT[ck][o], O padded 136->144 (zeros)
// ---------------------------------------------------------------------------
__global__ __launch_bounds__(256)
void wt_kernel(const float* __restrict__ w1, float* __restrict__ wT) {
  int i = blockIdx.x * blockDim.x + threadIdx.x;
  if (i >= CK * C1P) return;
  int ck = i / C1P, o = i - ck * C1P;
  wT[i] = (o < C1) ? w1[(size_t)o * CK + ck] : 0.f;
}

// ---------------------------------------------------------------------------
// Kernel B: fused Gram-tile (FP32 WMMA) + distance + per-row top-9
// One wave per 16 rows of one batch. A fragments held in registers
// (K-loop FULLY unrolled so aF[] has only static indices -> no scratch spill).
// ---------------------------------------------------------------------------
__global__ __launch_bounds__(32)
void dist_topk_kernel(const float* __restrict__ x2chan,
                      const float* __restrict__ sq,
                      int* __restrict__ kidx) {
  __shared__ float sdist[16][17];
  __shared__ float ssqRow[16];
  __shared__ float scol[16];
  const int b = blockIdx.y;
  const int rowBase = blockIdx.x * 16;
  const int lane = threadIdx.x;
  const int half = lane >> 4;          // A/B VGPR0 holds K0|K2, VGPR1 holds K1|K3
  const int m = lane & 15;
  const float* xb = x2chan + (size_t)b * C1 * NTOK;
  const float* arow = xb + (size_t)(2 * half) * NTOK + rowBase + m;

  // A fragments: 16 rows x 136 channels -> 34 K-steps of 4, coalesced loads
  v2f aF[34];
#pragma unroll
  for (int kk = 0; kk < 34; ++kk) {
    aF[kk].x = arow[(size_t)(4 * kk) * NTOK];
    aF[kk].y = arow[(size_t)(4 * kk + 1) * NTOK];
  }
  if (lane < 16) ssqRow[lane] = sq[b * NTOK + rowBase + lane];

  float td[KNN]; int ti[KNN];
#pragma unroll
  for (int s = 0; s < KNN; ++s) { td[s] = 3.0e38f; ti[s] = 0; }

  for (int ct = 0; ct < NTOK / 16; ++ct) {
    const int colBase = ct * 16;
    if (lane < 16) scol[lane] = sq[b * NTOK + colBase + lane];
    if (ct + 1 < NTOK / 16)
      __builtin_prefetch(&xb[colBase + 16 + m], 0, 1);   // global_prefetch_b8

    const float* bcol = xb + (size_t)(2 * half) * NTOK + colBase + m;
    v8f acc = {};
#pragma unroll                       // FULL unroll: keeps aF[] in VGPRs
    for (int kk = 0; kk < 34; ++kk) {
      v2f bF;
      bF.x = bcol[(size_t)(4 * kk) * NTOK];
      bF.y = bcol[(size_t)(4 * kk + 1) * NTOK];
      acc = __builtin_amdgcn_wmma_f32_16x16x4_f32(false, aF[kk], false, bF,
                                                  (short)0, acc, false, false);
    }
    __syncthreads();
    // D layout: VGPR v, lanes<16 -> M=v, N=lane ; lanes>=16 -> M=v+8, N=lane-16
    float sc = scol[m];
#pragma unroll
    for (int v = 0; v < 8; ++v) {
      int r = v + 8 * half;
      sdist[r][m] = ssqRow[r] + sc - 2.0f * acc[v];
    }
    __syncthreads();
    if (lane < 16) {
#pragma unroll
      for (int j = 0; j < 16; ++j) {
        float nd = sdist[lane][j];
        int ni = colBase + j;
        if (nd < td[KNN - 1]) {
#pragma unroll
          for (int s = 0; s < KNN; ++s) {   // register-resident sorted insert
            bool sw = nd < td[s];
            float odv = td[s]; int oiv = ti[s];
            td[s] = sw ? nd : odv; ti[s] = sw ? ni : oiv;
            nd = sw ? odv : nd;    ni = sw ? oiv : ni;
          }
        }
      }
    }
    __syncthreads();
  }
  if (lane < 16) {
#pragma unroll
    for (int s = 0; s < KNN; ++s)
      kidx[((size_t)(b * NTOK) + rowBase + lane) * KNN + s] = ti[s];
  }
}

// ---------------------------------------------------------------------------
// Kernel C: neighbor conv as GEMM  Y[o,n] = sum_ck wT[ck][o] * G[ck][n]
// G gathered into LDS per 16-token block; 9 waves, one 16x16 O-tile each.
// ---------------------------------------------------------------------------
__global__ __launch_bounds__(288)
void nnconv_kernel(const float* __restrict__ x2tok,
                   const int* __restrict__ kidx,
                   const float* __restrict__ wT,
                   const float* __restrict__ b1,
                   float* __restrict__ ychan) {       // [B][C1][N]
  __shared__ float g[48 * KNN * 16];                  // c-chunk of 48 -> 27.6 KB
  __shared__ int sidx[16][KNN];
  const int b = blockIdx.y;
  const int n0 = blockIdx.x * 16;
  const int tid = threadIdx.x;
  const int wave = tid >> 5;          // 0..8 -> O-tile
  const int lane = tid & 31;
  const int half = lane >> 4;
  const int lm = lane & 15;
  const int o0 = wave * 16;
  const float* xt = x2tok + (size_t)b * NTOK * C1;

  if (tid < 16 * KNN) {
    int t = tid / KNN, k = tid - t * KNN;
    sidx[t][k] = kidx[((size_t)(b * NTOK) + n0 + t) * KNN + k];
  }
  __syncthreads();

  v8f acc = {};
  const int cbase[3] = {0, 48, 96};
  for (int cc = 0; cc < 3; ++cc) {
    const int c0 = cbase[cc];
    const int nc = (cc == 2) ? 40 : 48;
    const int cb = nc >> 3;           // 8 contiguous channels per work item
    for (int p = tid; p < 144 * cb; p += 288) {
      int pair = p % 144;
      int cbi = p / 144;
      int t = pair / KNN, k = pair - t * KNN;
      int s = sidx[t][k];
      const float* src = xt + (size_t)s * C1 + c0 + cbi * 8;
#pragma unroll
      for (int q = 0; q < 8; ++q) {
        int lc = cbi * 8 + q;
        g[(lc * KNN + k) * 16 + t] = src[q];
      }
    }
    __syncthreads();
    const int ckn = nc * KNN;         // 432 or 360 (both % 8 == 0)
    const int ckg0 = c0 * KNN;
#pragma unroll 2
    for (int ckl = 0; ckl < ckn; ckl += 4) {
      int ckA = ckg0 + ckl + 2 * half;
      v2f a, bf;
      a.x = wT[(size_t)ckA * C1P + o0 + lm];          // coalesced (O padded)
      a.y = wT[(size_t)(ckA + 1) * C1P + o0 + lm];
      int ckB = ckl + 2 * half;
      bf.x = g[ckB * 16 + lm];
      bf.y = g[(ckB + 1) * 16 + lm];
      acc = __builtin_amdgcn_wmma_f32_16x16x4_f32(false, a, false, bf,
                                                  (short)0, acc, false, false);
    }
    __syncthreads();
  }
  const int n = n0 + lm;
#pragma unroll
  for (int v = 0; v < 8; ++v) {
    int o = o0 + v + 8 * half;
    if (o < C1)
      ychan[((size_t)(b * C1 + o)) * NTOK + n] = acc[v] + b1[o];
  }
}

// ---------------------------------------------------------------------------
// Kernel D: pixel-shuffle + pointwise conv (34 -> 32) + bias
// ---------------------------------------------------------------------------
__global__ __launch_bounds__(256)
void pw_kernel(const float* __restrict__ ychan,
               const float* __restrict__ pw_w,
               const float* __restrict__ pw_b,
               float* __restrict__ out) {
  __shared__ float spw[COUT * 34];
  __shared__ float spb[COUT];
  int tid = threadIdx.x;
  for (int i = tid; i < COUT * 34; i += 256) spw[i] = pw_w[i];
  if (tid < COUT) spb[tid] = pw_b[tid];
  __syncthreads();
  int gid = blockIdx.x * 256 + tid;           // over B*H*W = 131072
  int b = gid >> 12;
  int rem = gid & 4095;
  int h = rem >> 6, w = rem & 63;
  int sub = (h & 1) * 2 + (w & 1);
  int n = (h >> 1) * WS_ + (w >> 1);
  float v[34];
#pragma unroll
  for (int c = 0; c < 34; ++c)
    v[c] = ychan[((size_t)(b * C1 + c * 4 + sub)) * NTOK + n];
#pragma unroll
  for (int o = 0; o < COUT; ++o) {
    float s = spb[o];
#pragma unroll
    for (int c = 0; c < 34; ++c) s = fmaf(spw[o * 34 + c], v[c], s);
    out[(((size_t)(b * COUT + o)) * HH + h) * WW + w] = s;
  }
}

// ---------------------------------------------------------------------------
extern "C" void kernel_launch(void* const* d_in, const int* in_sizes, int n_in,
                              void* d_out, int out_size, void* d_ws, size_t ws_size,
                              hipStream_t stream) {
  const float* x    = (const float*)d_in[0];
  const float* w1   = (const float*)d_in[1];
  const float* b1   = (const float*)d_in[2];
  const float* pw_w = (const float*)d_in[3];
  const float* pw_b = (const float*)d_in[4];
  float* out = (float*)d_out;
  (void)in_sizes; (void)n_in; (void)out_size; (void)ws_size;

  char* ws = (char*)d_ws;
  size_t off = 0;
  auto carve = [&](size_t bytes) {
    void* p = ws + off;
    off = (off + bytes + 255) & ~(size_t)255;
    return p;
  };
  float* x2tok  = (float*)carve((size_t)BATCH * NTOK * C1 * 4);   // 17.8 MB
  float* x2chan = (float*)carve((size_t)BATCH * C1 * NTOK * 4);   // 17.8 MB
  float* sqv    = (float*)carve((size_t)BATCH * NTOK * 4);
  int*   kidx   = (int*)  carve((size_t)BATCH * NTOK * KNN * 4);
  float* wT     = (float*)carve((size_t)CK * C1P * 4);
  float* ychan  = (float*)carve((size_t)BATCH * C1 * NTOK * 4);   // 17.8 MB

  prep_kernel<<<dim3(NTOK / 256, BATCH), 256, 0, stream>>>(x, x2tok, x2chan, sqv);
  wt_kernel<<<(CK * C1P + 255) / 256, 256, 0, stream>>>(w1, wT);
  dist_topk_kernel<<<dim3(NTOK / 16, BATCH), 32, 0, stream>>>(x2chan, sqv, kidx);
  nnconv_kernel<<<dim3(NTOK / 16, BATCH), 288, 0, stream>>>(x2tok, kidx, wT, b1, ychan);
  pw_kernel<<<(BATCH * HH * WW + 255) / 256, 256, 0, stream>>>(ychan, pw_w, pw_b, out);
}